// TriangulationModule_72078141161691
// MI455X (gfx1250) — compile-verified
//
#include <hip/hip_runtime.h>
#include <hip/hip_bf16.h>

// MI455X / gfx1250, wave32. Triangulation via per-batch (10x20)x(20x4096) f32 GEMM
// on V_WMMA_F32_16X16X4_F32 for the A^T*A stage + per-lane Jacobi eigensolver.
// This revision makes the invalid-point test fully branchless so no EXEC-writing
// compare/saveexec code is scheduled into the WMMA window (WMMA requires EXEC==all-1s).

typedef __attribute__((ext_vector_type(2))) float v2f;
typedef __attribute__((ext_vector_type(8))) float v8f;

#define V_ 4
#define B_ 32
#define P_ 4096
#define TOL_ 1.001e-5f   // 1e-8 + 1e-5*1.0

__device__ __forceinline__ float xswap16(float x) {
  // half-wave swap (lane ^ 16) on wave32
  return __shfl_xor(x, 16, 32);
}

__global__ __launch_bounds__(256) void triangulate_wmma_kernel(
    const float* __restrict__ pts,    // (V,B,P,2)
    const float* __restrict__ proj,   // (V,B,3,4)
    float* __restrict__ out)          // (B,P,3)
{
  __shared__ float Cs[16 * 20];       // C_b, rows 10..15 zero-padded (M pad for WMMA)

  const int b   = blockIdx.y;
  const int tid = threadIdx.x;

  // ---- Stage C_b (16x20) into LDS: AtA(i,j) = sum_k C[m][k] * f[k] ----
  // entry order m: (0,0)(0,1)(0,2)(0,3)(1,1)(1,2)(1,3)(2,2)(2,3)(3,3)
  {
    const int Im[10] = {0,0,0,0,1,1,1,2,2,3};
    const int Jm[10] = {0,1,2,3,1,2,3,2,3,3};
    for (int t = tid; t < 16 * 20; t += 256) {
      int m = t / 20, k = t % 20;
      float cval = 0.0f;
      if (m < 10) {
        int i = Im[m], j = Jm[m];
        int v = k / 5, r = k % 5;
        const float* Pv = proj + (size_t)(v * B_ + b) * 12;
        float p1i = Pv[0 + i], p1j = Pv[0 + j];
        float p2i = Pv[4 + i], p2j = Pv[4 + j];
        float p3i = Pv[8 + i], p3j = Pv[8 + j];
        // features f = [x^2, x, y^2, y, 1] per view
        if (r == 0 || r == 2)  cval = p3i * p3j;
        else if (r == 1)       cval = -(p3i * p1j + p1i * p3j);
        else if (r == 3)       cval = -(p3i * p2j + p2i * p3j);
        else                   cval = p1i * p1j + p2i * p2j;
      }
      Cs[t] = cval;
    }
  }
  __syncthreads();

  const int  lane = tid & 31;
  const int  wave = tid >> 5;
  const bool hi   = (lane >= 16);
  const int  col  = lane & 15;
  const int  pb   = blockIdx.x * 256 + wave * 32;   // 32 points per wave

  // ---- A-operand slices (constant per block; 16x4 f32 layout per ISA 7.12.2) ----
  // lane<16: (M=lane, K=4s+0,4s+1); lane>=16: (M=lane-16, K=4s+2,4s+3)
  v2f Aop[5];
#pragma unroll
  for (int s = 0; s < 5; ++s) {
    const float* cp = &Cs[col * 20 + 4 * s + (hi ? 2 : 0)];
    v2f a; a[0] = cp[0]; a[1] = cp[1];
    Aop[s] = a;
  }

  // ---- Two 16-point tiles per wave: D = C_b * F  (AtA entries in D rows 0..9) ----
  v8f acc0; v8f acc1;
#pragma unroll
  for (int q = 0; q < 8; ++q) { acc0[q] = 0.0f; acc1[q] = 0.0f; }
  int myInvalid = 0;

#pragma unroll
  for (int tile = 0; tile < 2; ++tile) {
    const int p = pb + tile * 16 + col;     // B-operand column = lane&15
    float f[20];
    int   inv = 0;                          // branchless accumulation (no && / ||)
#pragma unroll
    for (int v = 0; v < V_; ++v) {
      const float* q = pts + ((size_t)(v * B_ + b) * P_ + p) * 2;
      float x = q[0], y = q[1];
      f[5 * v + 0] = x * x;
      f[5 * v + 1] = x;
      f[5 * v + 2] = y * y;
      f[5 * v + 3] = y;
      f[5 * v + 4] = 1.0f;
      int bx = (fabsf(x + 1.0f) <= TOL_) ? 1 : 0;
      int by = (fabsf(y + 1.0f) <= TOL_) ? 1 : 0;
      inv = inv | (bx & by);
    }
    // lane "owns" point pb+lane: tile0 for lanes<16, tile1 for lanes>=16
    const bool own = (tile == 0) ? !hi : hi;
    myInvalid = own ? inv : myInvalid;

    v8f acc;
#pragma unroll
    for (int q = 0; q < 8; ++q) acc[q] = 0.0f;
#pragma unroll
    for (int s = 0; s < 5; ++s) {
      // B-operand 4x16 f32: lanes<16 hold K=4s+0,4s+1; lanes>=16 hold K=4s+2,4s+3
      v2f bop;
      bop[0] = hi ? f[4 * s + 2] : f[4 * s + 0];
      bop[1] = hi ? f[4 * s + 3] : f[4 * s + 1];
      acc = __builtin_amdgcn_wmma_f32_16x16x4_f32(
          /*neg_a=*/false, Aop[s], /*neg_b=*/false, bop,
          /*c_mod=*/(short)0, acc, /*reuse_a=*/false, /*reuse_b=*/false);
    }
    if (tile == 0) acc0 = acc; else acc1 = acc;
  }

  // ---- Redistribute D so every lane holds all 10 AtA entries of point pb+lane ----
  // D layout: v_j = row j (lanes<16) / row j+8 (lanes>=16), column = lane&15.
  float e[10];
#pragma unroll
  for (int r = 0; r < 8; ++r) {
    float swz = xswap16(acc1[r]);           // hi lanes receive tile1 rows 0..7
    e[r] = hi ? swz : acc0[r];
  }
  {
    float s8 = xswap16(acc0[0]);            // lo lanes receive tile0 rows 8,9
    float s9 = xswap16(acc0[1]);
    e[8] = hi ? acc1[0] : s8;
    e[9] = hi ? acc1[1] : s9;
  }

  // ---- Per-lane symmetric 4x4 eigensolve (cyclic Jacobi, 6 sweeps) ----
  float m[4][4];
  m[0][0] = e[0]; m[0][1] = m[1][0] = e[1]; m[0][2] = m[2][0] = e[2]; m[0][3] = m[3][0] = e[3];
  m[1][1] = e[4]; m[1][2] = m[2][1] = e[5]; m[1][3] = m[3][1] = e[6];
  m[2][2] = e[7]; m[2][3] = m[3][2] = e[8]; m[3][3] = e[9];
  float vv[4][4] = {{1,0,0,0},{0,1,0,0},{0,0,1,0},{0,0,0,1}};

  const int PPr[6] = {0,0,0,1,1,2};
  const int QQr[6] = {1,2,3,2,3,3};
  for (int sweep = 0; sweep < 6; ++sweep) {
#pragma unroll
    for (int pr = 0; pr < 6; ++pr) {
      const int pi = PPr[pr], qi = QQr[pr];
      float apq = m[pi][qi];
      float app = m[pi][pi], aqq = m[qi][qi];
      bool  rot = fabsf(apq) > 1e-30f;
      float tau = (aqq - app) / (2.0f * (rot ? apq : 1.0f));
      float t   = copysignf(1.0f, tau) / (fabsf(tau) + sqrtf(1.0f + tau * tau));
      float c   = 1.0f / sqrtf(1.0f + t * t);
      float s   = t * c;
      c = rot ? c : 1.0f;
      s = rot ? s : 0.0f;
#pragma unroll
      for (int k = 0; k < 4; ++k) {   // M = M * J
        float t1 = m[k][pi], t2 = m[k][qi];
        m[k][pi] = c * t1 - s * t2;
        m[k][qi] = s * t1 + c * t2;
      }
#pragma unroll
      for (int k = 0; k < 4; ++k) {   // M = J^T * M
        float t1 = m[pi][k], t2 = m[qi][k];
        m[pi][k] = c * t1 - s * t2;
        m[qi][k] = s * t1 + c * t2;
      }
#pragma unroll
      for (int k = 0; k < 4; ++k) {   // V = V * J
        float t1 = vv[k][pi], t2 = vv[k][qi];
        vv[k][pi] = c * t1 - s * t2;
        vv[k][qi] = s * t1 + c * t2;
      }
    }
  }

  // eigenvector of smallest eigenvalue (branchless argmin over diagonal)
  float d  = m[0][0];
  float h0 = vv[0][0], h1 = vv[1][0], h2 = vv[2][0], h3 = vv[3][0];
#pragma unroll
  for (int k = 1; k < 4; ++k) {
    bool better = m[k][k] < d;
    d  = better ? m[k][k]  : d;
    h0 = better ? vv[0][k] : h0;
    h1 = better ? vv[1][k] : h1;
    h2 = better ? vv[2][k] : h2;
    h3 = better ? vv[3][k] : h3;
  }

  // ---- Homogenize, clip, sentinel. Global rescale is provably identity:
  //      outputs are clipped to [-10,10] and the sentinel is (0,0,5), so
  //      max_abs <= 10 and scale == 1 always. ----
  bool  nz = (h3 != 0.0f);
  float sw = nz ? h3 : 1.0f;
  float x  = nz ? h0 / sw : h0;
  float y  = nz ? h1 / sw : h1;
  float z  = nz ? h2 / sw : h2;
  x = fminf(fmaxf(x, -10.0f), 10.0f);
  y = fminf(fmaxf(y, -10.0f), 10.0f);
  z = fminf(fmaxf(z, -10.0f), 10.0f);
  x = myInvalid ? 0.0f : x;
  y = myInvalid ? 0.0f : y;
  z = myInvalid ? 5.0f : z;

  const int p = pb + lane;
  float* o = out + ((size_t)b * P_ + p) * 3;
  o[0] = x; o[1] = y; o[2] = z;
}

extern "C" void kernel_launch(void* const* d_in, const int* in_sizes, int n_in,
                              void* d_out, int out_size, void* d_ws, size_t ws_size,
                              hipStream_t stream) {
  const float* pts  = (const float*)d_in[0];   // (V,B,P,2) f32
  const float* proj = (const float*)d_in[1];   // (V,B,3,4) f32
  float* out = (float*)d_out;                  // (B,P,3) f32
  dim3 grid(P_ / 256, B_);                     // 512 blocks
  dim3 block(256);                             // 8 waves/block, 32 pts/wave
  hipLaunchKernelGGL(triangulate_wmma_kernel, grid, block, 0, stream, pts, proj, out);
}